// ExpLogDiceLoss_58128087384124
// MI455X (gfx1250) — compile-verified
//
#include <hip/hip_runtime.h>
#include <hip/hip_bf16.h>

// ExpLogDiceLoss for N rows x C=32 classes, single streaming pass.
//
// Math: log_softmax => probs p in (0,1]; npr = log(sum_rows p[:,c]),
// ni = log(sum_{label==c} p[:,c]), logaddexp(log_ngt,npr) = log(max(cnt,1)+S_all).
// loss = mean over present classes of pow(-(LG2 + ni - log(cnt+S_all)), 0.3).
//
// Wave layout: 16-row x 32-class tile in V_WMMA_F32_16X16X4_F32 A-matrix layout:
//   lane p: row m = p&15, half h = p>>4, holds classes c with ((c>>1)&1)==h
//   (8 aligned float2 pairs per lane). Row sums Z_m for all 16 rows come from
//   8 chained v_wmma_f32_16x16x4_f32 (B = ones, C-accumulated) in full f32.

#define NCLS 32

typedef float v2f __attribute__((ext_vector_type(2)));
typedef float v8f __attribute__((ext_vector_type(8)));

__global__ void __launch_bounds__(256, 2)
dice_partial_kernel(const float* __restrict__ score,
                    const int* __restrict__ label,
                    float* __restrict__ ws,   // [0:32)=S_all [32:64)=S_lab [64:96)=cnt
                    int n_tiles) {
    __shared__ float sAll[NCLS], sLab[NCLS], sCnt[NCLS];
    const int tid = threadIdx.x;
    if (tid < NCLS) { sAll[tid] = 0.0f; sLab[tid] = 0.0f; sCnt[tid] = 0.0f; }
    __syncthreads();

    const int lane = tid & 31;
    const int m    = lane & 15;   // row within tile
    const int h    = lane >> 4;   // class-half: this lane owns classes with bit1==h
    const int gwave  = blockIdx.x * (blockDim.x >> 5) + (tid >> 5);
    const int nwaves = gridDim.x * (blockDim.x >> 5);

    float accAll[16];
#pragma unroll
    for (int j = 0; j < 16; ++j) accAll[j] = 0.0f;

    const v2f ones = {1.0f, 1.0f};

    for (int t = gwave; t < n_tiles; t += nwaves) {
        const int    r0      = t << 4;
        const size_t rowBase = (size_t)(r0 + m) * NCLS;

        // ---- load 16 f32 (8 x b64 pairs): classes 4j'+2h, 4j'+2h+1 ----
        float e[16];
#pragma unroll
        for (int jp = 0; jp < 8; ++jp) {
            v2f p2 = *(const v2f*)(score + rowBase + jp * 4 + 2 * h);
            e[2 * jp]     = p2.x;
            e[2 * jp + 1] = p2.y;
        }

        // prefetch this wave's next tile (global_prefetch_b8)
        if (t + nwaves < n_tiles) {
            __builtin_prefetch(score + rowBase + (size_t)nwaves * 16 * NCLS, 0, 3);
        }

        // ---- row max: 16 local values + complementary half via shfl_xor(16) ----
        float mx = e[0];
#pragma unroll
        for (int j = 1; j < 16; ++j) mx = fmaxf(mx, e[j]);
        mx = fmaxf(mx, __shfl_xor(mx, 16));

#pragma unroll
        for (int j = 0; j < 16; ++j) e[j] = __expf(e[j] - mx);

        // ---- Z_m for 16 rows: 8 chained f32 WMMAs, D[m,n] = sum_c e[m,c] ----
        v8f zacc = {0.f, 0.f, 0.f, 0.f, 0.f, 0.f, 0.f, 0.f};
#pragma unroll
        for (int jp = 0; jp < 8; ++jp) {
            v2f a = {e[2 * jp], e[2 * jp + 1]};   // A vgpr0=K(2h), vgpr1=K(2h+1)
            zacc = __builtin_amdgcn_wmma_f32_16x16x4_f32(
                false, a, false, ones, (short)0, zacc, false, false);
        }

        // ---- pull Z_m back to lane p (D: vgpr v, lane l -> Z_{v + 8*(l>>4)}) ----
        const int sel = lane & 7;
        float z = zacc[0];
#pragma unroll
        for (int v = 1; v < 8; ++v) z = (sel == v) ? zacc[v] : z;
        const float zsw = __shfl_xor(z, 16);
        const int   b3  = (lane >> 3) & 1;
        z = (b3 == h) ? z : zsw;                  // now z = Z_{lane&15}
        const float rz = __builtin_amdgcn_rcpf(z);   // single v_rcp_f32 (~22-bit)

        // ---- per-class S_all accumulation (lane-local, 16 fmas) ----
#pragma unroll
        for (int j = 0; j < 16; ++j) accAll[j] = fmaf(e[j], rz, accAll[j]);

        // ---- label commit: each row hits exactly one (lane, slot) ----
        const int lab   = label[r0 + m];
        const int j_lab = ((lab >> 2) << 1) | (lab & 1);  // slot if this half owns it
        float v = e[0];
#pragma unroll
        for (int j = 1; j < 16; ++j) v = (j_lab == j) ? e[j] : v;
        if (((lab >> 1) & 1) == h) {              // owner half commits once per row
            atomicAdd(&sLab[lab], v * rz);
            atomicAdd(&sCnt[lab], 1.0f);
        }
    }

    // ---- block reduce S_all via LDS float atomics (ds_add_f32) ----
#pragma unroll
    for (int j = 0; j < 16; ++j) {
        const int cls = ((j >> 1) << 2) + 2 * h + (j & 1);
        atomicAdd(&sAll[cls], accAll[j]);
    }
    __syncthreads();

    if (tid < NCLS) {
        atomicAdd(&ws[tid],            sAll[tid]);
        atomicAdd(&ws[NCLS + tid],     sLab[tid]);
        atomicAdd(&ws[2 * NCLS + tid], sCnt[tid]);
    }
}

__global__ void dice_init_kernel(float* __restrict__ ws) {
    const int i = threadIdx.x;
    if (i < 3 * NCLS) ws[i] = 0.0f;
}

__global__ void dice_final_kernel(const float* __restrict__ ws, float* __restrict__ out) {
    const int c = threadIdx.x;                       // 32 threads, one wave
    const float sAll = ws[c];
    const float sLab = ws[NCLS + c];
    const float cnt  = ws[2 * NCLS + c];
    const bool present = cnt > 0.0f;

    // log_dice = LG2 + log(S_lab) - log(max(cnt,1) + S_all)
    float loss = 0.0f, np = 0.0f;
    if (present) {
        const float log_dice = 0.6931471805599453f + logf(sLab) - logf(fmaxf(cnt, 1.0f) + sAll);
        loss = powf(-log_dice, 0.3f);
        np   = 1.0f;
    }
#pragma unroll
    for (int off = 16; off > 0; off >>= 1) {
        loss += __shfl_down(loss, off);
        np   += __shfl_down(np, off);
    }
    if (c == 0) out[0] = loss / np;   // LOSS_WEIGHT = 1.0
}

extern "C" void kernel_launch(void* const* d_in, const int* in_sizes, int n_in,
                              void* d_out, int out_size, void* d_ws, size_t ws_size,
                              hipStream_t stream) {
    const float* score = (const float*)d_in[0];   // [N, 32] f32
    const int*   label = (const int*)d_in[1];     // [N] int
    float*       ws    = (float*)d_ws;            // >= 96 floats
    float*       out   = (float*)d_out;           // 1 float

    const int n_rows  = in_sizes[0] / NCLS;
    const int n_tiles = n_rows >> 4;              // 16 rows per tile (N is a multiple of 16)

    dice_init_kernel<<<1, 128, 0, stream>>>(ws);
    dice_partial_kernel<<<1024, 256, 0, stream>>>(score, label, ws, n_tiles);
    dice_final_kernel<<<1, 32, 0, stream>>>(ws, out);
}